// MSE_DQ_FK_68075231642107
// MI455X (gfx1250) — compile-verified
//
#include <hip/hip_runtime.h>
#include <hip/hip_bf16.h>
#include <math.h>

#define NB 64
#define NT 1024
#define NJ 22
#define NC 176   // J*8

// PARENTS, EE mask (4,8,13,17,21), NS mask (not sparse, not root)
static constexpr int PAR[NJ]  = {0,0,1,2,3,0,5,6,7,0,9,10,11,12,11,14,15,16,11,18,19,20};
static constexpr int ISEE[NJ] = {0,0,0,0,1,0,0,0,1,0,0,0,0,1,0,0,0,1,0,0,0,1};
static constexpr int ISNS[NJ] = {0,1,1,1,0,1,1,1,0,1,1,1,1,0,1,1,1,0,1,1,1,0};

typedef __attribute__((ext_vector_type(2))) float v2f;
typedef __attribute__((ext_vector_type(8))) float v8f;

__device__ __forceinline__ void qmul(const float a[4], const float b[4], float d[4]) {
  d[0] = a[0]*b[0] - a[1]*b[1] - a[2]*b[2] - a[3]*b[3];
  d[1] = a[0]*b[1] + a[1]*b[0] + a[2]*b[3] - a[3]*b[2];
  d[2] = a[0]*b[2] - a[1]*b[3] + a[2]*b[0] + a[3]*b[1];
  d[3] = a[0]*b[3] + a[1]*b[2] - a[2]*b[1] + a[3]*b[0];
}

__device__ __forceinline__ void qmul_conj_a(const float a[4], const float b[4], float d[4]) {
  const float c[4] = {a[0], -a[1], -a[2], -a[3]};
  qmul(c, b, d);
}

__device__ __forceinline__ void qnormalize(float q[4]) {
  const float n = q[0]*q[0] + q[1]*q[1] + q[2]*q[2] + q[3]*q[3];
  const float inv = 1.0f / sqrtf(n);
  q[0] *= inv; q[1] *= inv; q[2] *= inv; q[3] *= inv;
}

// d = v + w*t + cross(qv, t), t = 2*cross(qv, v)
__device__ __forceinline__ void qrot(const float q[4], const float v[3], float d[3]) {
  const float tx = 2.0f * (q[2]*v[2] - q[3]*v[1]);
  const float ty = 2.0f * (q[3]*v[0] - q[1]*v[2]);
  const float tz = 2.0f * (q[1]*v[1] - q[2]*v[0]);
  d[0] = v[0] + q[0]*tx + (q[2]*tz - q[3]*ty);
  d[1] = v[1] + q[0]*ty + (q[3]*tx - q[1]*tz);
  d[2] = v[2] + q[0]*tz + (q[1]*ty - q[2]*tx);
}

__device__ __forceinline__ void qmat(const float q[4], float m[9]) {
  const float w = q[0], x = q[1], y = q[2], z = q[3];
  m[0] = 1.0f - 2.0f*(y*y + z*z); m[1] = 2.0f*(x*y - w*z);          m[2] = 2.0f*(x*z + w*y);
  m[3] = 2.0f*(x*y + w*z);        m[4] = 1.0f - 2.0f*(x*x + z*z);   m[5] = 2.0f*(y*z - w*x);
  m[6] = 2.0f*(x*z - w*y);        m[7] = 2.0f*(y*z + w*x);          m[8] = 1.0f - 2.0f*(x*x + y*y);
}

// -------------------- main kernel: one pose (b,t) per thread --------------------
__global__ __launch_bounds__(256)
void fk_loss_kernel(const float* __restrict__ ikp,  const float* __restrict__ decp,
                    const float* __restrict__ tgtp, const float* __restrict__ meanp,
                    const float* __restrict__ stdp, const float* __restrict__ offp,
                    float* __restrict__ ws)
{
  const int gid = blockIdx.x * 256 + threadIdx.x;     // 0..65535 exactly
  const int b = gid >> 10;                            // / NT
  const int t = gid & (NT - 1);
  const long baseF = (long)b * (NC * NT) + t;         // dec/tgt: + c*NT
  const long baseI = (long)b * ((NC - 8) * NT) + t;   // ik: channel (c-8)

  // Raw (normalized) quats per joint + FK global rot/pos per source.
  // Fully-unrolled constant indexing => SSA values; liveness keeps only
  // the active ancestor chain.
  float qr_ik[NJ][4], qr_de[NJ][4], qr_tg[NJ][4];
  float ro_ik[NJ][4], ro_de[NJ][4], ro_tg[NJ][4];
  float po_ik[NJ][3], po_de[NJ][3], po_tg[NJ][3];

  // roots: ik/dec root dq forced to identity; tgt root comes from data
  qr_ik[0][0] = 1.f; qr_ik[0][1] = 0.f; qr_ik[0][2] = 0.f; qr_ik[0][3] = 0.f;
  qr_de[0][0] = 1.f; qr_de[0][1] = 0.f; qr_de[0][2] = 0.f; qr_de[0][3] = 0.f;
#pragma unroll
  for (int k = 0; k < 4; ++k)
    qr_tg[0][k] = tgtp[baseF + (long)k * NT] * stdp[k] + meanp[k];
  qnormalize(qr_tg[0]);

  // FK root: global rot = identity (local[0] forced identity), pos = 0 for all sources
#pragma unroll
  for (int k = 0; k < 4; ++k) {
    const float v = (k == 0) ? 1.f : 0.f;
    ro_ik[0][k] = v; ro_de[0][k] = v; ro_tg[0][k] = v;
  }
#pragma unroll
  for (int k = 0; k < 3; ++k) { po_ik[0][k] = 0.f; po_de[0][k] = 0.f; po_tg[0][k] = 0.f; }

  float see_p = 0.f, see_m = 0.f, sns_p = 0.f, sns_m = 0.f;

#pragma unroll
  for (int j = 1; j < NJ; ++j) {
    const int p = PAR[j];

    // --- load + denorm the 4 real-quat channels for each source (coalesced in t) ---
    float a_ik[4], a_de[4], a_tg[4];
#pragma unroll
    for (int k = 0; k < 4; ++k) {
      const int c = j * 8 + k;
      const float mu = meanp[c];   // lane-uniform -> s_load
      const float sd = stdp[c];
      a_ik[k] = ikp [baseI + (long)(c - 8) * NT] * sd + mu;
      a_de[k] = decp[baseF + (long)c * NT]       * sd + mu;
      a_tg[k] = tgtp[baseF + (long)c * NT]       * sd + mu;
    }
    qnormalize(a_ik); qnormalize(a_de); qnormalize(a_tg);
#pragma unroll
    for (int k = 0; k < 4; ++k) { qr_ik[j][k] = a_ik[k]; qr_de[j][k] = a_de[k]; qr_tg[j][k] = a_tg[k]; }

    // --- local rotation: conj(q[parent]) * q[j] ---
    float l_ik[4], l_de[4], l_tg[4];
    qmul_conj_a(qr_ik[p], a_ik, l_ik);
    qmul_conj_a(qr_de[p], a_de, l_de);
    qmul_conj_a(qr_tg[p], a_tg, l_tg);

    // --- FK ---
    qmul(ro_ik[p], l_ik, ro_ik[j]);
    qmul(ro_de[p], l_de, ro_de[j]);
    qmul(ro_tg[p], l_tg, ro_tg[j]);

    const float off[3] = { offp[j*3 + 0], offp[j*3 + 1], offp[j*3 + 2] };  // uniform -> s_load
    float r0[3], r1[3], r2[3];
    qrot(ro_ik[p], off, r0);
    qrot(ro_de[p], off, r1);
    qrot(ro_tg[p], off, r2);
#pragma unroll
    for (int k = 0; k < 3; ++k) {
      po_ik[j][k] = r0[k] + po_ik[p][k];
      po_de[j][k] = r1[k] + po_de[p][k];
      po_tg[j][k] = r2[k] + po_tg[p][k];
    }

    // --- loss contributions (branches fold after unroll) ---
    if (ISEE[j]) {  // ik vs tgt
#pragma unroll
      for (int k = 0; k < 3; ++k) { const float d = po_ik[j][k] - po_tg[j][k]; see_p += d * d; }
      float m0[9], m1[9];
      qmat(ro_ik[j], m0); qmat(ro_tg[j], m1);
#pragma unroll
      for (int k = 0; k < 9; ++k) { const float d = m0[k] - m1[k]; see_m += d * d; }
    }
    if (ISNS[j]) {  // dec vs ik
#pragma unroll
      for (int k = 0; k < 3; ++k) { const float d = po_de[j][k] - po_ik[j][k]; sns_p += d * d; }
      float m0[9], m1[9];
      qmat(ro_de[j], m0); qmat(ro_ik[j], m1);
#pragma unroll
      for (int k = 0; k < 9; ++k) { const float d = m0[k] - m1[k]; sns_m += d * d; }
    }
  }

  // per-pose loss with mse denominators folded in (final /(B*T) in reduce kernel):
  // ee: 5 joints * {3 pos, 9 rm};  ns: 16 joints * {3 pos, 9 rm}, weight 0.1
  float loss = see_p * (1.0f / 15.0f) + see_m * (1.0f / 45.0f)
             + 0.1f * (sns_p * (1.0f / 48.0f) + sns_m * (1.0f / 144.0f));

  __shared__ float sm[256];
  sm[threadIdx.x] = loss;
  __syncthreads();
#pragma unroll
  for (int s = 128; s > 0; s >>= 1) {
    if (threadIdx.x < s) sm[threadIdx.x] += sm[threadIdx.x + s];
    __syncthreads();
  }
  if (threadIdx.x == 0) ws[blockIdx.x] = sm[0];
}

// -------------------- WMMA reduction: 256 partials -> scalar --------------------
// D = sum_c A_c * Ones + C ; with B = ones, D[m,n] = rowsum(m). Column 0 lives in
// lane 0 (M=0..7, vgpr 0..7) and lane 16 (M=8..15). total = sum(lane0) + sum(lane16).
__global__ __launch_bounds__(32)
void reduce_wmma_kernel(const float* __restrict__ ws, float* __restrict__ out)
{
  const int lane = threadIdx.x;  // one full wave32, EXEC all ones
#if __has_builtin(__builtin_amdgcn_wmma_f32_16x16x4_f32)
  v8f acc = {0.f, 0.f, 0.f, 0.f, 0.f, 0.f, 0.f, 0.f};
  v2f ones; ones.x = 1.0f; ones.y = 1.0f;
#pragma unroll
  for (int c = 0; c < 4; ++c) {
    v2f a;
    a.x = ws[c * 64 + lane];        // A vgpr0
    a.y = ws[c * 64 + 32 + lane];   // A vgpr1
    acc = __builtin_amdgcn_wmma_f32_16x16x4_f32(
        /*neg_a=*/false, a, /*neg_b=*/false, ones,
        /*c_mod=*/(short)0, acc, /*reuse_a=*/false, /*reuse_b=*/false);
  }
  float s = acc[0] + acc[1] + acc[2] + acc[3] + acc[4] + acc[5] + acc[6] + acc[7];
  __shared__ float sm[32];
  sm[lane] = s;
  __syncthreads();
  if (lane == 0) out[0] = (sm[0] + sm[16]) * (1.0f / (float)(NB * NT));
#else
  __shared__ float sm[32];
  float s = 0.f;
  for (int i = lane; i < 256; i += 32) s += ws[i];
  sm[lane] = s;
  __syncthreads();
  if (lane == 0) {
    float tot = 0.f;
    for (int i = 0; i < 32; ++i) tot += sm[i];
    out[0] = tot * (1.0f / (float)(NB * NT));
  }
#endif
}

extern "C" void kernel_launch(void* const* d_in, const int* in_sizes, int n_in,
                              void* d_out, int out_size, void* d_ws, size_t ws_size,
                              hipStream_t stream)
{
  (void)in_sizes; (void)n_in; (void)out_size; (void)ws_size;
  // setup_inputs order: input(0, UNUSED), input_ik(1), input_decoder(2),
  //                     target(3), mean_dqs(4), std_dqs(5), offsets(6)
  const float* ikp  = (const float*)d_in[1];
  const float* decp = (const float*)d_in[2];
  const float* tgtp = (const float*)d_in[3];
  const float* mean = (const float*)d_in[4];
  const float* stdv = (const float*)d_in[5];
  const float* off  = (const float*)d_in[6];
  float* ws  = (float*)d_ws;    // 256 block partials
  float* out = (float*)d_out;

  fk_loss_kernel<<<256, 256, 0, stream>>>(ikp, decp, tgtp, mean, stdv, off, ws);
  reduce_wmma_kernel<<<1, 32, 0, stream>>>(ws, out);
}